// GlobalPNAModel_52828097741406
// MI455X (gfx1250) — compile-verified
//
#include <hip/hip_runtime.h>
#include <math.h>

typedef float v2f __attribute__((ext_vector_type(2)));
typedef float v8f __attribute__((ext_vector_type(8)));

#define NN   1048576
#define DD   128
#define GG   4096
#define HID  50
#define HPAD 64
#define GDIM 201
#define TAIL 33
#define OUTW 234   // GDIM + TAIL

#define SELU_ALPHA 1.6732632423543772f
#define SELU_SCALE 1.0507009873554805f

// ---------------------------------------------------------------- bounds ----
// starts[g] = lower_bound(batch, g); starts[GG] = NN (batch values < GG).
__global__ void k_bounds(const int* __restrict__ batch, int* __restrict__ starts) {
    int g = blockIdx.x * blockDim.x + threadIdx.x;
    if (g > GG) return;
    int lo = 0, hi = NN;
    while (lo < hi) {
        int mid = (lo + hi) >> 1;
        if (batch[mid] < g) lo = mid + 1; else hi = mid;
    }
    starts[g] = lo;
}

// ------------------------------------------------------------- aggregate ----
// One block per group. 256 threads = 8 row-slots x 32 lanes; each wave loads
// one full 512B row as float4 per lane (perfectly coalesced b128 loads).
__global__ void k_aggregate(const float* __restrict__ x,
                            const int* __restrict__ starts,
                            float* __restrict__ aggr) {
    const int g    = blockIdx.x;
    const int tid  = threadIdx.x;
    const int lane = tid & 31;   // column group: cols 4*lane..4*lane+3
    const int slot = tid >> 5;   // 0..7 row slot (one wave per slot)
    const int start = starts[g];
    const int end   = starts[g + 1];

    float s0 = 0.f, s1 = 0.f, s2 = 0.f, s3 = 0.f;
    float q0 = 0.f, q1 = 0.f, q2 = 0.f, q3 = 0.f;
    float mx0 = -INFINITY, mx1 = -INFINITY, mx2 = -INFINITY, mx3 = -INFINITY;
    float mn0 =  INFINITY, mn1 =  INFINITY, mn2 =  INFINITY, mn3 =  INFINITY;

    for (int r = start + slot; r < end; r += 8) {
        const float4 v = *(const float4*)(x + (size_t)r * DD + lane * 4);
        __builtin_prefetch(x + (size_t)(r + 16) * DD + lane * 4, 0, 0);
        s0 += v.x; s1 += v.y; s2 += v.z; s3 += v.w;
        q0 += v.x * v.x; q1 += v.y * v.y; q2 += v.z * v.z; q3 += v.w * v.w;
        mx0 = fmaxf(mx0, v.x); mx1 = fmaxf(mx1, v.y);
        mx2 = fmaxf(mx2, v.z); mx3 = fmaxf(mx3, v.w);
        mn0 = fminf(mn0, v.x); mn1 = fminf(mn1, v.y);
        mn2 = fminf(mn2, v.z); mn3 = fminf(mn3, v.w);
    }

    __shared__ float4 ls[256], lq[256], lmx[256], lmn[256];  // 16 KB of 320 KB
    ls[tid]  = make_float4(s0, s1, s2, s3);
    lq[tid]  = make_float4(q0, q1, q2, q3);
    lmx[tid] = make_float4(mx0, mx1, mx2, mx3);
    lmn[tid] = make_float4(mn0, mn1, mn2, mn3);
    __syncthreads();

    for (int off = 128; off >= 32; off >>= 1) {
        if (tid < off) {
            float4 a, b;
            a = ls[tid];  b = ls[tid + off];
            ls[tid]  = make_float4(a.x + b.x, a.y + b.y, a.z + b.z, a.w + b.w);
            a = lq[tid];  b = lq[tid + off];
            lq[tid]  = make_float4(a.x + b.x, a.y + b.y, a.z + b.z, a.w + b.w);
            a = lmx[tid]; b = lmx[tid + off];
            lmx[tid] = make_float4(fmaxf(a.x,b.x), fmaxf(a.y,b.y), fmaxf(a.z,b.z), fmaxf(a.w,b.w));
            a = lmn[tid]; b = lmn[tid + off];
            lmn[tid] = make_float4(fminf(a.x,b.x), fminf(a.y,b.y), fminf(a.z,b.z), fminf(a.w,b.w));
        }
        __syncthreads();
    }

    if (tid < 32) {
        const float cnt = fmaxf((float)(end - start), 1.0f);
        const float inv = 1.0f / cnt;
        float4 s = ls[tid], q = lq[tid];
        float4 mean = make_float4(s.x * inv, s.y * inv, s.z * inv, s.w * inv);
        float4 var  = make_float4(q.x * inv - mean.x * mean.x,
                                  q.y * inv - mean.y * mean.y,
                                  q.z * inv - mean.z * mean.z,
                                  q.w * inv - mean.w * mean.w);
        float4 sd = make_float4(sqrtf(fmaxf(var.x, 0.f) + 1e-5f),
                                sqrtf(fmaxf(var.y, 0.f) + 1e-5f),
                                sqrtf(fmaxf(var.z, 0.f) + 1e-5f),
                                sqrtf(fmaxf(var.w, 0.f) + 1e-5f));
        float* base = aggr + (size_t)g * (4 * DD);
        ((float4*)(base          ))[tid] = mean;
        ((float4*)(base + 1 * DD))[tid] = sd;
        ((float4*)(base + 2 * DD))[tid] = lmx[tid];
        ((float4*)(base + 3 * DD))[tid] = lmn[tid];
    }
}

// ------------------------------------------------------------ gemm1 wmma ----
// hpre[4096 x 64pad] = aggr[4096 x 512] @ W1[512 x 50] + b1, via f32 WMMA.
// Block = 128 threads = 4 waves; wave w computes the 16x16 tile at N = 16*w.
// Out-of-range N is handled by clamped-address loads + v_cndmask zeroing so
// the inner loop stays branch-free (no exec-mask churn around the WMMAs).
__global__ void k_gemm1_wmma(const float* __restrict__ aggr,
                             const float* __restrict__ W1,
                             const float* __restrict__ bias1,
                             float* __restrict__ hpre) {
    const int m0   = blockIdx.x * 16;
    const int tid  = threadIdx.x;
    const int wave = tid >> 5;
    const int l    = tid & 31;
    const int l16  = l & 15;
    const int hi   = l >> 4;           // 0: K pair {0,1}, 1: K pair {2,3}
    const int n    = wave * 16 + l16;
    const bool nok = (n < HID);
    const int  nc  = nok ? n : 0;      // clamped column -> always-valid address

    const float* arow = aggr + (size_t)(m0 + l16) * 512 + hi * 2;
    const float* bcol = W1 + nc + (size_t)(hi * 2) * HID;

    v8f acc = {};
    for (int k0 = 0; k0 < 512; k0 += 4) {
        v2f a;
        a.x = arow[k0];
        a.y = arow[k0 + 1];
        const float bx = bcol[(size_t)k0 * HID];
        const float by = bcol[(size_t)k0 * HID + HID];
        v2f b;
        b.x = nok ? bx : 0.f;          // v_cndmask, not an exec branch
        b.y = nok ? by : 0.f;
        acc = __builtin_amdgcn_wmma_f32_16x16x4_f32(
            /*neg_a=*/false, a, /*neg_b=*/false, b,
            /*c_mod=*/(short)0, acc, /*reuse_a=*/false, /*reuse_b=*/false);
    }

    if (nok) {
        const float bb = bias1[n];
#pragma unroll
        for (int v = 0; v < 8; ++v) {
            // C/D layout: VGPR v holds row (v + 8*hi), col n.
            hpre[(size_t)(m0 + v + hi * 8) * HPAD + n] = acc[v] + bb;
        }
    }
}

// ----------------------------------------------------- selu + LN + tail ----
// One block (64 threads) per group: selu, LayerNorm over 50, zero-pad h to 64,
// and copy the 33-wide tail of u into the output.
__global__ void k_act_ln_tail(const float* __restrict__ hpre,
                              const float* __restrict__ ln_g,
                              const float* __restrict__ ln_b,
                              const float* __restrict__ u,
                              float* __restrict__ h,
                              float* __restrict__ out) {
    const int g   = blockIdx.x;
    const int tid = threadIdx.x;   // 64
    __shared__ float red[64];

    float s = 0.f;
    if (tid < HID) {
        const float xv = hpre[(size_t)g * HPAD + tid];
        s = SELU_SCALE * (xv > 0.f ? xv : SELU_ALPHA * (expf(xv) - 1.f));
    }

    red[tid] = s;
    __syncthreads();
    for (int off = 32; off > 0; off >>= 1) {
        if (tid < off) red[tid] += red[tid + off];
        __syncthreads();
    }
    const float mu = red[0] * (1.0f / HID);
    __syncthreads();

    const float d = (tid < HID) ? (s - mu) : 0.f;
    red[tid] = d * d;
    __syncthreads();
    for (int off = 32; off > 0; off >>= 1) {
        if (tid < off) red[tid] += red[tid + off];
        __syncthreads();
    }
    const float var = red[0] * (1.0f / HID);
    const float rs  = rsqrtf(var + 1e-5f);

    float hv = 0.f;
    if (tid < HID) hv = d * rs * ln_g[tid] + ln_b[tid];
    h[(size_t)g * HPAD + tid] = hv;   // cols 50..63 zeroed -> K-pad for gemm2

    if (tid < TAIL)
        out[(size_t)g * OUTW + GDIM + tid] = u[(size_t)g * 64 + (64 - TAIL) + tid];
}

// ------------------------------------------------------------ gemm2 wmma ----
// out[:, :201] = h[4096 x 50(pad 64)] @ W2[50 x 201] + b2.  K padded to 52
// (h cols 50..51 are zero); N tiled 13x16.  Guards are clamp+select so the
// WMMA loop is branch-free.
__global__ void k_gemm2_wmma(const float* __restrict__ h,
                             const float* __restrict__ W2,
                             const float* __restrict__ bias2,
                             float* __restrict__ out) {
    const int m0  = blockIdx.x * 16;
    const int n0  = blockIdx.y * 16;   // 13 tiles -> up to 208
    const int l   = threadIdx.x;
    const int l16 = l & 15;
    const int hi  = l >> 4;
    const int n   = n0 + l16;
    const bool nok = (n < GDIM);
    const int  nc  = nok ? n : 0;

    const float* arow = h + (size_t)(m0 + l16) * HPAD + hi * 2;

    v8f acc = {};
    for (int k0 = 0; k0 < 52; k0 += 4) {
        v2f a;
        a.x = arow[k0];          // padded columns (50..63) are zero
        a.y = arow[k0 + 1];
        const int k   = k0 + hi * 2;
        const int kc0 = (k     < HID) ? k     : 0;   // clamped -> valid address
        const int kc1 = (k + 1 < HID) ? k + 1 : 0;
        const float bx = W2[(size_t)kc0 * GDIM + nc];
        const float by = W2[(size_t)kc1 * GDIM + nc];
        v2f b;
        b.x = (nok && k     < HID) ? bx : 0.f;       // v_cndmask
        b.y = (nok && k + 1 < HID) ? by : 0.f;
        acc = __builtin_amdgcn_wmma_f32_16x16x4_f32(
            false, a, false, b, (short)0, acc, false, false);
    }

    if (nok) {
        const float bb = bias2[n];
#pragma unroll
        for (int v = 0; v < 8; ++v)
            out[(size_t)(m0 + v + hi * 8) * OUTW + n] = acc[v] + bb;
    }
}

// ----------------------------------------------------------------- launch ---
extern "C" void kernel_launch(void* const* d_in, const int* in_sizes, int n_in,
                              void* d_out, int out_size, void* d_ws, size_t ws_size,
                              hipStream_t stream) {
    const float* x     = (const float*)d_in[0];
    // d_in[1] edge_index, d_in[2] edge_attr: unused by the reference
    const float* u     = (const float*)d_in[3];
    const int*   batch = (const int*)  d_in[4];
    const float* W1    = (const float*)d_in[5];
    const float* b1    = (const float*)d_in[6];
    const float* lng   = (const float*)d_in[7];
    const float* lnb   = (const float*)d_in[8];
    const float* W2    = (const float*)d_in[9];
    const float* b2    = (const float*)d_in[10];
    float* out = (float*)d_out;

    char* ws = (char*)d_ws;
    int*   starts = (int*)ws;                                   // (G+1) ints
    float* aggr   = (float*)(ws + 32768);                       // 4096*512 f32 = 8 MB
    float* hpre   = (float*)(ws + 32768 + (size_t)GG*512*4);    // 4096*64 f32 = 1 MB
    float* hbuf   = (float*)(ws + 32768 + (size_t)GG*512*4 + (size_t)GG*HPAD*4);

    k_bounds     <<<(GG + 1 + 127) / 128, 128, 0, stream>>>(batch, starts);
    k_aggregate  <<<GG, 256, 0, stream>>>(x, starts, aggr);
    k_gemm1_wmma <<<GG / 16, 128, 0, stream>>>(aggr, W1, b1, hpre);
    k_act_ln_tail<<<GG, 64, 0, stream>>>(hpre, lng, lnb, u, hbuf, out);
    k_gemm2_wmma <<<dim3(GG / 16, 13), 32, 0, stream>>>(hbuf, W2, b2, out);
}